// NAND_Exhaustive_64433099375168
// MI455X (gfx1250) — compile-verified
//
#include <hip/hip_runtime.h>
#include <hip/hip_bf16.h>

typedef __attribute__((ext_vector_type(2))) float v2f;
typedef __attribute__((ext_vector_type(8))) float v8f;

#define VEC    128
#define NV     8
#define NROWS  256
#define BATCHN 512
#define DIMS   1024   // NV * VEC
#define EPSF   1e-8f

// One block = 16 (batch) x 16 (rows) tile. 8 waves; wave n handles sub-vector n.
// Phase 1: WMMA f32 16x16x4 chain over K=128 -> dot tile; norms folded into same loads.
// Phase 2: exhaustive 256-combo product expansion, 4 combos per thread per pair,
//          contiguous float4 stores (output-bandwidth-bound: 134 MB writes).
__global__ __launch_bounds__(256) void nand_exhaustive_kernel(
    const float* __restrict__ q, const float* __restrict__ w,
    float* __restrict__ out) {
  __shared__ float cs_lds[NV][16][16];

  const int tid  = threadIdx.x;
  const int wv   = tid >> 5;       // sub-vector index n = wave id (0..7)
  const int lane = tid & 31;
  const int half = lane >> 4;      // K-half selector per the 32-bit A/B layout
  const int lm   = lane & 15;      // M (for A) / N (for B) index
  const int b0 = blockIdx.x * 16;
  const int r0 = blockIdx.y * 16;

  // A frag (16x4 f32): lane L -> (M = L&15, K = k0 + 2*(L>>4) + {0,1})
  // B frag (4x16 f32): lane L -> (N = L&15, K = k0 + 2*(L>>4) + {0,1})
  const float* ap = q + (size_t)(b0 + lm) * DIMS + wv * VEC + half * 2;
  const float* bp = w + (size_t)(r0 + lm) * DIMS + wv * VEC + half * 2;

  v8f acc = {};
  float qs = 0.0f, ws = 0.0f;
#pragma unroll 4
  for (int k = 0; k < VEC / 4; ++k) {
    v2f a = *(const v2f*)ap; ap += 4;
    v2f b = *(const v2f*)bp; bp += 4;
    qs = fmaf(a.x, a.x, fmaf(a.y, a.y, qs));
    ws = fmaf(b.x, b.x, fmaf(b.y, b.y, ws));
    // 8 args: (neg_a, A, neg_b, B, c_mod, C, reuse_a, reuse_b)
    acc = __builtin_amdgcn_wmma_f32_16x16x4_f32(
        false, a, false, b, (short)0, acc, false, false);
  }

  // Each lane holds half the K range for row (lm); fold halves across lane^16.
  qs += __shfl_xor(qs, 16, 32);
  ws += __shfl_xor(ws, 16, 32);
  const float qn = fmaxf(sqrtf(qs), EPSF);  // lane L holds qn[M = L&15]
  const float wn = fmaxf(sqrtf(ws), EPSF);  // lane L holds wn[N = L&15]

  // C/D layout: VGPR j, lane L -> (M = j + 8*(L>>4), N = L&15)
#pragma unroll
  for (int j = 0; j < 8; ++j) {
    const float qnM = __shfl(qn, j + (half << 3), 32);
    const float csv = fmaxf(acc[j] / (qnM * wn), 0.0f);
    cs_lds[wv][j + (half << 3)][lm] = csv;
  }
  __syncthreads();

  // Phase 2: thread t covers combos [4*(t&63), +4) of pair (t>>6) in each group.
  // combos[i,j] = 1 if bit j of i is 0 -> term_j = bit ? (1-cs_j) : cs_j.
  const int cgrp = tid & 63;
  const int psub = tid >> 6;
  const int base = cgrp << 2;     // low 2 bits of combo vary within the float4
#pragma unroll 2
  for (int it = 0; it < 64; ++it) {
    const int pair = (it << 2) + psub;   // 0..255
    const int m = pair >> 4;
    const int r = pair & 15;
    float c[8];
#pragma unroll
    for (int n = 0; n < 8; ++n) c[n] = cs_lds[n][m][r];

    // Shared product over bits 2..7 (fixed across the 4 combos).
    float high = ((base >> 2) & 1) ? (1.0f - c[2]) : c[2];
#pragma unroll
    for (int j = 3; j < 8; ++j)
      high *= ((base >> j) & 1) ? (1.0f - c[j]) : c[j];

    const float c0 = c[0], c1 = c[1];
    const float n0 = 1.0f - c0, n1 = 1.0f - c1;
    float4 o;
    o.x = high * (c0 * c1);   // bits (0,0)
    o.y = high * (n0 * c1);   // bits (1,0)
    o.z = high * (c0 * n1);   // bits (0,1)
    o.w = high * (n0 * n1);   // bits (1,1)

    const size_t off =
        (((size_t)(b0 + m) * NROWS) + (size_t)(r0 + r)) * 256 + (size_t)base;
    *(float4*)(out + off) = o;
  }
}

extern "C" void kernel_launch(void* const* d_in, const int* in_sizes, int n_in,
                              void* d_out, int out_size, void* d_ws, size_t ws_size,
                              hipStream_t stream) {
  (void)in_sizes; (void)n_in; (void)out_size; (void)d_ws; (void)ws_size;
  const float* q = (const float*)d_in[0];   // (512, 1024) f32
  const float* w = (const float*)d_in[1];   // (256, 1024) f32
  float* out = (float*)d_out;               // (512, 256, 256) f32
  dim3 grid(BATCHN / 16, NROWS / 16);
  nand_exhaustive_kernel<<<grid, 256, 0, stream>>>(q, w, out);
}